// EmbeddingGroup_72456098284168
// MI455X (gfx1250) — compile-verified
//
#include <hip/hip_runtime.h>
#include <math.h>

// ---------------------------------------------------------------------------
// VQ-VAE quantize + 1x1 conv for gfx1250 (MI455X), wave32, V_WMMA_F32_16X16X4_F32
// ---------------------------------------------------------------------------

#define BATCH    16
#define E_DIM    1024
#define N_E      256
#define HW       1024          // 32*32
#define N_PIX    16384         // BATCH*HW
#define BETA     0.25f

typedef float v2f __attribute__((ext_vector_type(2)));
typedef float v8f __attribute__((ext_vector_type(8)));

// workspace layout (in floats)
#define WS_EMBT   0            // embT[E_DIM][N_E]  = 262144 floats
#define WS_HE2    262144       // 0.5*||emb_e||^2   = 256 floats
#define WS_HIST   262400       // histogram, 256 ints
#define WS_LOSS   262656       // 1 float (sum of squared diffs)
#define WS_IDX    262657       // min_idx as int, 16384 ints

// d_out layout (floats): out | loss | perplexity | min_idx
#define OUT_LOSS  16777216
#define OUT_PERP  16777217
#define OUT_IDX   16777218

// ---------------------------------------------------------------------------
// Kernel 0: transpose codebook to embT[c][e], compute 0.5*||e||^2,
//           zero histogram and loss accumulator (re-done every launch so the
//           whole pipeline is deterministic under graph replay).
// ---------------------------------------------------------------------------
__global__ __launch_bounds__(256) void vq_prep(const float* __restrict__ emb,
                                               float* __restrict__ ws) {
  __shared__ float red[256];
  const int e = blockIdx.x;       // 0..255
  const int t = threadIdx.x;      // 0..255
  float* embT = ws + WS_EMBT;
  float local = 0.f;
  for (int c = t; c < E_DIM; c += 256) {
    float v = emb[e * E_DIM + c];
    embT[c * N_E + e] = v;
    local += v * v;
  }
  red[t] = local;
  __syncthreads();
  for (int s = 128; s > 0; s >>= 1) {
    if (t < s) red[t] += red[t + s];
    __syncthreads();
  }
  if (t == 0) ws[WS_HE2 + e] = 0.5f * red[0];
  if (e == 0) {
    ((int*)(ws + WS_HIST))[t] = 0;
    if (t == 0) ws[WS_LOSS] = 0.f;
  }
}

// ---------------------------------------------------------------------------
// Kernel 1: per 16-pixel tile: WMMA distance GEMM vs all 256 codes, argmin
//           (tie-break = lowest index, matching jnp.argmin), histogram,
//           and the (z_q - z)^2 loss partial sum.
//           Block = 128 threads (4 waves); wave w covers codes [w*64, w*64+64).
// ---------------------------------------------------------------------------
__global__ __launch_bounds__(128) void vq_argmin(const float* __restrict__ z,
                                                 float* __restrict__ ws,
                                                 float* __restrict__ out) {
  const int r    = blockIdx.x;          // pixel tile 0..1023
  const int tid  = threadIdx.x;
  const int lane = tid & 31;
  const int wave = tid >> 5;
  const int half = lane >> 4;           // 0 / 1
  const int lm   = lane & 15;
  const int kk   = half << 1;           // K offset 0 or 2 inside the x4 step

  const int b   = r >> 6;               // batch
  const int hw0 = (r & 63) << 4;        // spatial offset of pixel 0 in tile
  const float* zb   = z + (size_t)b * E_DIM * HW + hw0;   // zb[c*HW + m]
  const float* embT = ws + WS_EMBT;
  const float* he2  = ws + WS_HE2;

  v8f acc[4];
#pragma unroll
  for (int t = 0; t < 4; ++t)
#pragma unroll
    for (int j = 0; j < 8; ++j) acc[t][j] = 0.f;

  const int ebase = wave * 64;

  for (int kb = 0; kb < E_DIM / 4; ++kb) {
    const int c0 = (kb << 2) + kk;
    // A fragment: 16 pixels x 4 channels, lanes 0-15 = rows, halves = K pairs
    v2f a;
    a.x = zb[(size_t)c0 * HW + lm];
    a.y = zb[(size_t)(c0 + 1) * HW + lm];
    const float* bt0 = embT + (size_t)c0 * N_E;
    const float* bt1 = bt0 + N_E;
#pragma unroll
    for (int t = 0; t < 4; ++t) {
      const int ec = ebase + (t << 4) + lm;
      v2f bv;
      bv.x = bt0[ec];
      bv.y = bt1[ec];
      acc[t] = __builtin_amdgcn_wmma_f32_16x16x4_f32(
          false, a, false, bv, (short)0, acc[t], false, false);
    }
  }

  // score = 0.5*||e||^2 - z.e  (|| z ||^2 dropped: constant per pixel)
  float bestv[8];
  int   beste[8];
#pragma unroll
  for (int j = 0; j < 8; ++j) { bestv[j] = 3.4e38f; beste[j] = 0x7fffffff; }
#pragma unroll
  for (int t = 0; t < 4; ++t) {
    const int e = ebase + (t << 4) + lm;
    const float he = he2[e];
#pragma unroll
    for (int j = 0; j < 8; ++j) {
      const float v = he - acc[t][j];
      if (v < bestv[j]) { bestv[j] = v; beste[j] = e; }   // strict < keeps lowest e
    }
  }

  // argmin across the 16 lanes of each half-wave (codes), keep lowest index on ties
#pragma unroll
  for (int j = 0; j < 8; ++j) {
    float v = bestv[j];
    int   e = beste[j];
#pragma unroll
    for (int m = 8; m >= 1; m >>= 1) {
      const float ov = __shfl_xor(v, m, 32);
      const int   oe = __shfl_xor(e, m, 32);
      if (ov < v || (ov == v && oe < e)) { v = ov; e = oe; }
    }
    bestv[j] = v; beste[j] = e;
  }

  __shared__ float sv[4][16];
  __shared__ int   se[4][16];
  __shared__ int   idx16[16];
  __shared__ float wred[4];

  if (lm == 0) {
#pragma unroll
    for (int j = 0; j < 8; ++j) {
      sv[wave][j + half * 8] = bestv[j];
      se[wave][j + half * 8] = beste[j];
    }
  }
  __syncthreads();

  if (tid < 16) {
    float v = sv[0][tid];
    int   e = se[0][tid];
#pragma unroll
    for (int w = 1; w < 4; ++w) {
      const float ov = sv[w][tid];
      const int   oe = se[w][tid];
      if (ov < v || (ov == v && oe < e)) { v = ov; e = oe; }
    }
    idx16[tid] = e;
    ((int*)(ws + WS_IDX))[r * 16 + tid] = e;
    out[OUT_IDX + r * 16 + tid] = (float)e;      // tuple concat promotes to f32
    atomicAdd((int*)(ws + WS_HIST) + e, 1);
  }
  __syncthreads();

  // loss partial: sum over tile of (emb[idx][c] - z)^2 ; consecutive threads ->
  // consecutive pixels (coalesced z), embT row is 1KB and L1-resident.
  float local = 0.f;
  for (int k = tid; k < 16 * E_DIM; k += 128) {
    const int m = k & 15;
    const int c = k >> 4;
    const float zv = zb[(size_t)c * HW + m];
    const float q  = embT[(size_t)c * N_E + idx16[m]];
    const float d  = q - zv;
    local += d * d;
  }
#pragma unroll
  for (int m = 16; m >= 1; m >>= 1) local += __shfl_xor(local, m, 32);
  if (lane == 0) wred[wave] = local;
  __syncthreads();
  if (tid == 0) atomicAdd(ws + WS_LOSS, wred[0] + wred[1] + wred[2] + wred[3]);
}

// ---------------------------------------------------------------------------
// Kernel 2: out[b][o][hw] = sum_c emb[idx][c] * conv_w[o][c] + conv_b[o]
//           M = output channel, N = pixel -> coalesced 64B stores.
//           Block = 256 threads (8 waves), each wave owns 4 o-tiles (64 chans),
//           grid.y = 2 halves of the 1024 output channels.
// ---------------------------------------------------------------------------
__global__ __launch_bounds__(256) void vq_conv(const float* __restrict__ w,
                                               const float* __restrict__ bias,
                                               const float* __restrict__ ws,
                                               float* __restrict__ out) {
  const int r    = blockIdx.x;
  const int tid  = threadIdx.x;
  const int lane = tid & 31;
  const int wave = tid >> 5;
  const int half = lane >> 4;
  const int lm   = lane & 15;
  const int kk   = half << 1;

  const int b   = r >> 6;
  const int hw0 = (r & 63) << 4;
  const float* embT = ws + WS_EMBT;
  const int qidx = ((const int*)(ws + WS_IDX))[r * 16 + lm];  // pixel lm's code
  const int obase = blockIdx.y * 512 + wave * 64;

  v8f acc[4];
#pragma unroll
  for (int t = 0; t < 4; ++t)
#pragma unroll
    for (int j = 0; j < 8; ++j) acc[t][j] = 0.f;

  for (int kb = 0; kb < E_DIM / 4; ++kb) {
    const int c0 = (kb << 2) + kk;
    // B fragment: z_q (K x 16 pixels) via transposed-codebook gather
    v2f bv;
    bv.x = embT[(size_t)c0 * N_E + qidx];
    bv.y = embT[(size_t)(c0 + 1) * N_E + qidx];
#pragma unroll
    for (int t = 0; t < 4; ++t) {
      const int orow = obase + (t << 4) + lm;
      v2f a;
      a.x = w[(size_t)orow * E_DIM + c0];
      a.y = w[(size_t)orow * E_DIM + c0 + 1];
      acc[t] = __builtin_amdgcn_wmma_f32_16x16x4_f32(
          false, a, false, bv, (short)0, acc[t], false, false);
    }
  }

  float* ob = out + (size_t)b * E_DIM * HW + hw0;
#pragma unroll
  for (int t = 0; t < 4; ++t)
#pragma unroll
    for (int j = 0; j < 8; ++j) {
      const int o = obase + (t << 4) + j + half * 8;
      ob[(size_t)o * HW + lm] = acc[t][j] + bias[o];
    }
}

// ---------------------------------------------------------------------------
// Kernel 3: finalize scalars: loss = (1+beta)*sum/N_elems ; perplexity.
// ---------------------------------------------------------------------------
__global__ __launch_bounds__(256) void vq_final(const float* __restrict__ ws,
                                                float* __restrict__ out) {
  __shared__ float red[256];
  const int t = threadIdx.x;
  const float p = (float)((const int*)(ws + WS_HIST))[t] / (float)N_PIX;
  red[t] = p * logf(p + 1e-10f);
  __syncthreads();
  for (int s = 128; s > 0; s >>= 1) {
    if (t < s) red[t] += red[t + s];
    __syncthreads();
  }
  if (t == 0) {
    out[OUT_PERP] = expf(-red[0]);
    out[OUT_LOSS] = ws[WS_LOSS] * (1.0f + BETA) / (float)(N_PIX * E_DIM);
  }
}

// ---------------------------------------------------------------------------
extern "C" void kernel_launch(void* const* d_in, const int* in_sizes, int n_in,
                              void* d_out, int out_size, void* d_ws, size_t ws_size,
                              hipStream_t stream) {
  (void)in_sizes; (void)n_in; (void)out_size; (void)ws_size;
  const float* z    = (const float*)d_in[0];
  const float* emb  = (const float*)d_in[1];
  const float* cw   = (const float*)d_in[2];
  const float* cb   = (const float*)d_in[3];
  float* out = (float*)d_out;
  float* ws  = (float*)d_ws;

  vq_prep  <<<dim3(N_E),     dim3(256), 0, stream>>>(emb, ws);
  vq_argmin<<<dim3(1024),    dim3(128), 0, stream>>>(z, ws, out);
  vq_conv  <<<dim3(1024, 2), dim3(256), 0, stream>>>(cw, cb, ws, out);
  vq_final <<<dim3(1),       dim3(256), 0, stream>>>(ws, out);
}